// GNNExtrapolation_14654428414226
// MI455X (gfx1250) — compile-verified
//
#include <hip/hip_runtime.h>

typedef __attribute__((ext_vector_type(2))) float v2f;
typedef __attribute__((ext_vector_type(8))) float v8f;

#define NNODES   10000
#define T_IN     12
#define T_OUT    12
#define T_TOT    24
#define BATCH    32
#define CH       2
#define K1       9
#define NH       4
#define KDIM     48          // T_IN * NH
#define ROWS     64          // BATCH*CH rows per node tile (b*2+c)

__device__ __forceinline__ float selu_f(float v) {
    const float scale = 1.0507009873554805f;
    const float alpha = 1.6732632423543772f;
    return scale * (v > 0.0f ? v : alpha * (__expf(v) - 1.0f));
}

// ---------------- sigma = max(nearest_dists) ----------------
__global__ void init_max_kernel(unsigned int* bits) {
    if (threadIdx.x == 0 && blockIdx.x == 0) bits[0] = 0u;  // dists are >= 0
}

__global__ void max_reduce_kernel(const float* __restrict__ d, int n,
                                  unsigned int* __restrict__ bits) {
    __shared__ float s[256];
    float m = 0.0f;
    for (int i = blockIdx.x * blockDim.x + threadIdx.x; i < n;
         i += gridDim.x * blockDim.x)
        m = fmaxf(m, d[i]);
    s[threadIdx.x] = m;
    __syncthreads();
    for (int off = 128; off > 0; off >>= 1) {
        if ((int)threadIdx.x < off)
            s[threadIdx.x] = fmaxf(s[threadIdx.x], s[threadIdx.x + off]);
        __syncthreads();
    }
    if (threadIdx.x == 0)
        atomicMax(bits, __float_as_uint(s[0]));   // bit-order == float order for >=0
}

// ---------------- out[:, 0:12] = x ----------------
__global__ void copy_x_kernel(const float* __restrict__ x, float* __restrict__ out) {
    const int total = BATCH * T_IN * NNODES * CH;           // 7,680,000
    int i = blockIdx.x * blockDim.x + threadIdx.x;
    if (i >= total) return;
    const int per_b = T_IN * NNODES * CH;
    int b = i / per_b;
    int r = i - b * per_b;
    out[(size_t)b * (T_TOT * NNODES * CH) + r] = x[i];
}

// ---------------- fused aggregate + WMMA GEMM + SELU ----------------
// grid = NNODES blocks, block = 64 threads (2 wave32)
__global__ void gnn_main_kernel(const float* __restrict__ x,
                                const int*   __restrict__ nn,
                                const float* __restrict__ nd,
                                const float* __restrict__ W,
                                const float* __restrict__ bias,
                                const unsigned int* __restrict__ sig_bits,
                                float* __restrict__ out) {
    __shared__ float s_w[K1][NH];
    __shared__ int   s_idx[K1];
    __shared__ float s_feat[ROWS][KDIM];

    const int n   = blockIdx.x;
    const int tid = threadIdx.x;

    // sigma from the reduction kernel
    const float sig_max = __uint_as_float(sig_bits[0]);
    const float sigma   = sig_max * (1.0f / 500.0f);
    const float inv_s2  = 1.0f / (sigma * sigma);

    if (tid < K1) {
        int v = nn[n * K1 + tid];
        s_idx[tid] = (v < 0) ? 0 : v;           // clamp; weight zeroed below
    }
    if (tid < K1 * NH) {
        const int k = tid / NH, h = tid % NH;
        const float d   = nd[n * K1 + k];
        const float lam = (float)(h + 1) * 0.25f;
        float wv = __expf(-d * d * lam * inv_s2);
        if (nn[n * K1 + k] == -1 || wv < 1e-8f) wv = 0.0f;
        s_w[k][h] = wv;
    }
    __syncthreads();

    // ---- per-thread feature row (48 values): row r = b*CH + c ----
    {
        const int r = tid;
        const int b = r >> 1;
        const int c = r & 1;
        for (int t = 0; t < T_IN; ++t) {
            const float* xp = x + (((size_t)b * T_IN + t) * NNODES) * CH + c;
            float a0 = 0.f, a1 = 0.f, a2 = 0.f, a3 = 0.f;
#pragma unroll
            for (int k = 0; k < K1; ++k) {
                const float xv = xp[(size_t)s_idx[k] * CH];
                a0 += xv * s_w[k][0];
                a1 += xv * s_w[k][1];
                a2 += xv * s_w[k][2];
                a3 += xv * s_w[k][3];
            }
            s_feat[r][t * NH + 0] = a0;
            s_feat[r][t * NH + 1] = a1;
            s_feat[r][t * NH + 2] = a2;
            s_feat[r][t * NH + 3] = a3;
        }
    }
    __syncthreads();

    // ---- 64x48 @ 48x12 via V_WMMA_F32_16X16X4_F32; 2 row-tiles per wave ----
    const int wave = tid >> 5;
    const int lane = tid & 31;
    const int half = lane >> 4;       // 0: lanes 0-15, 1: lanes 16-31
    const int col  = lane & 15;       // N index in B/C/D layouts
    const int colc = (col < T_OUT) ? col : (T_OUT - 1);   // safe clamp
    const bool cok = (col < T_OUT);

    // Preload all 12 B fragments into registers, branch-free:
    // unconditional (clamped) loads + value select, no exec masking.
    v2f bf[12];
#pragma unroll
    for (int s = 0; s < 12; ++s) {
        const int kb = s * 4 + 2 * half;     // K rows this lane supplies
        const float w0 = W[(size_t)kb * T_OUT + colc];
        const float w1 = W[(size_t)(kb + 1) * T_OUT + colc];
        bf[s].x = cok ? w0 : 0.0f;
        bf[s].y = cok ? w1 : 0.0f;
    }

    v8f acc0 = {};
    v8f acc1 = {};
#pragma unroll
    for (int s = 0; s < 12; ++s) {
        const int kb = s * 4 + 2 * half;
        // A fragments from LDS: lane row = tile_base + (lane&15), K = kb, kb+1
        v2f a0 = *(const v2f*)&s_feat[wave * 32 + col][kb];
        v2f a1 = *(const v2f*)&s_feat[wave * 32 + 16 + col][kb];
        acc0 = __builtin_amdgcn_wmma_f32_16x16x4_f32(
            false, a0, false, bf[s], (short)0, acc0, false, false);
        acc1 = __builtin_amdgcn_wmma_f32_16x16x4_f32(
            false, a1, false, bf[s], (short)0, acc1, false, false);
    }

    // ---- SELU + scatter to out[b, 12+col, n, c] ----
    const float bbv = bias[colc];
    const float bb  = cok ? bbv : 0.0f;
    if (cok) {
#pragma unroll
        for (int i = 0; i < 8; ++i) {
            const int M  = i + 8 * half;                 // C/D VGPR layout
            const int r0 = wave * 32 + M;
            const int r1 = wave * 32 + 16 + M;
            const int b0 = r0 >> 1, c0 = r0 & 1;
            const int b1 = r1 >> 1, c1 = r1 & 1;
            out[(((size_t)b0 * T_TOT + T_IN + col) * NNODES + n) * CH + c0] =
                selu_f(acc0[i] + bb);
            out[(((size_t)b1 * T_TOT + T_IN + col) * NNODES + n) * CH + c1] =
                selu_f(acc1[i] + bb);
        }
    }
}

extern "C" void kernel_launch(void* const* d_in, const int* in_sizes, int n_in,
                              void* d_out, int out_size, void* d_ws, size_t ws_size,
                              hipStream_t stream) {
    const float* x  = (const float*)d_in[0];   // (32,12,10000,2)
    const int*   nn = (const int*)d_in[1];     // (10000,9)
    const float* nd = (const float*)d_in[2];   // (10000,9)
    const float* W  = (const float*)d_in[3];   // (48,12)
    const float* b  = (const float*)d_in[4];   // (12,)
    float* out = (float*)d_out;                // (32,24,10000,2)

    unsigned int* sig_bits = (unsigned int*)d_ws;
    const int n_dists = in_sizes[2];           // 90000

    init_max_kernel<<<1, 32, 0, stream>>>(sig_bits);
    max_reduce_kernel<<<64, 256, 0, stream>>>(nd, n_dists, sig_bits);

    {
        const int total = BATCH * T_IN * NNODES * CH;
        copy_x_kernel<<<(total + 255) / 256, 256, 0, stream>>>(x, out);
    }

    gnn_main_kernel<<<NNODES, ROWS, 0, stream>>>(x, nn, nd, W, b, sig_bits, out);
}